// HybridGNNLayer_17506286698856
// MI455X (gfx1250) — compile-verified
//
#include <hip/hip_runtime.h>
#include <hip/hip_bf16.h>

#define N_NODES 20000
#define N_EDGES 640000
#define IN_DIM  128
#define OUT_DIM 256
#define ED_DIM  32
#define NT_NUM  16
#define H_NUM   8
#define C_DIM   16
#define GATC    128
#define GINEC   128

typedef __attribute__((ext_vector_type(2))) float v2f;
typedef __attribute__((ext_vector_type(8))) float v8f;

// ---------- helpers ----------
__device__ __forceinline__ unsigned f2ord(float f) {
    unsigned u = __float_as_uint(f);
    return (u & 0x80000000u) ? ~u : (u | 0x80000000u);
}
__device__ __forceinline__ float ord2f(unsigned u) {
    return (u & 0x80000000u) ? __uint_as_float(u & 0x7FFFFFFFu) : __uint_as_float(~u);
}

// Async global->LDS 16B copy (GLOBAL_LOAD_ASYNC_TO_LDS_B128, ASYNCcnt-tracked).
// lds_off: byte offset within the workgroup LDS segment (per-lane VDST),
// gaddr:   64-bit global byte address (per-lane VADDR), GV mode.
__device__ __forceinline__ void async_ld_b128(unsigned lds_off, const void* gaddr) {
    asm volatile("global_load_async_to_lds_b128 %0, %1, off"
                 :: "v"(lds_off), "v"((unsigned long long)(uintptr_t)gaddr)
                 : "memory");
}
__device__ __forceinline__ void async_wait0() {
    asm volatile("s_wait_asynccnt 0x0" ::: "memory");
}

// ---------- kernel 1: precompute small tables ----------
// Wae[32][8]     : gat_edge_W folded with a_edge
// type_ae[16][8] : gat_type_emb @ gat_edge_W folded with a_edge
// type_ep[16][128]: gine_type_emb @ gine_edge_W
// cb2[256]       : comb_b + gat_b @ comb_W[0:128,:]   (bias folding for k_final)
__global__ __launch_bounds__(256) void k_precompute(
    const float* __restrict__ gat_edge_W, const float* __restrict__ a_edge,
    const float* __restrict__ gat_type_emb, const float* __restrict__ gine_type_emb,
    const float* __restrict__ gine_edge_W,
    const float* __restrict__ gat_b, const float* __restrict__ comb_W,
    const float* __restrict__ comb_b,
    float* __restrict__ Wae, float* __restrict__ type_ae, float* __restrict__ type_ep,
    float* __restrict__ cb2)
{
    __shared__ float sWae[ED_DIM * H_NUM];
    int tid = threadIdx.x;
    {   // Wae[k][h] = sum_c gat_edge_W[k][h*16+c] * a_edge[h*16+c]
        int k = tid >> 3, hh = tid & 7;
        float v = 0.f;
        #pragma unroll
        for (int c = 0; c < C_DIM; ++c)
            v += gat_edge_W[k * 128 + hh * C_DIM + c] * a_edge[hh * C_DIM + c];
        sWae[tid] = v;
        Wae[tid] = v;
    }
    __syncthreads();
    if (tid < NT_NUM * H_NUM) {
        int t = tid >> 3, hh = tid & 7;
        float v = 0.f;
        #pragma unroll
        for (int k = 0; k < ED_DIM; ++k)
            v += gat_type_emb[t * ED_DIM + k] * sWae[k * H_NUM + hh];
        type_ae[tid] = v;
    }
    for (int i = tid; i < NT_NUM * IN_DIM; i += 256) {
        int t = i >> 7, j = i & 127;
        float v = 0.f;
        #pragma unroll
        for (int k = 0; k < ED_DIM; ++k)
            v += gine_type_emb[t * ED_DIM + k] * gine_edge_W[k * IN_DIM + j];
        type_ep[i] = v;
    }
    {   // cb2[j] = comb_b[j] + sum_k gat_b[k] * comb_W[k][j]
        float v = comb_b[tid];
        for (int k = 0; k < GATC; ++k)
            v += gat_b[k] * comb_W[k * OUT_DIM + tid];
        cb2[tid] = v;
    }
}

// ---------- kernel 2: zero-init accumulators ----------
__global__ __launch_bounds__(256) void k_init(
    unsigned* __restrict__ mx, float* __restrict__ denom,
    float* __restrict__ acc1, float* __restrict__ acc2, int nmx, int nacc)
{
    int i = blockIdx.x * blockDim.x + threadIdx.x;
    int stride = gridDim.x * blockDim.x;
    for (int j = i; j < nmx; j += stride) { mx[j] = 0u; denom[j] = 0.f; }
    for (int j = i; j < nacc; j += stride) { acc1[j] = 0.f; acc2[j] = 0.f; }
}

// ---------- kernel 3: h = x @ gat_W  (N x 128 @ 128 x 128), WMMA f32 16x16x4 ----------
__global__ __launch_bounds__(256) void k_gemm_h(
    const float* __restrict__ x, const float* __restrict__ W, float* __restrict__ h)
{
    __shared__ float As[16 * 128];           // 8 KB, contiguous tile in global
    int r0 = blockIdx.x * 16;
    int tid = threadIdx.x;
    {
        unsigned lb = (unsigned)(uintptr_t)As;
        const char* gb = (const char*)(x + (size_t)r0 * 128);
        async_ld_b128(lb + tid * 16, gb + tid * 16);
        async_ld_b128(lb + 4096 + tid * 16, gb + 4096 + tid * 16);
        async_wait0();
    }
    __syncthreads();
    int wave = tid >> 5, lane = tid & 31;
    int half = lane >> 4, l16 = lane & 15, kb = half * 2;
    int n0 = wave * 16;
    v8f acc = {};
    #pragma unroll 4
    for (int k = 0; k < 128; k += 4) {
        v2f a, b;
        a.x = As[l16 * 128 + k + kb];
        a.y = As[l16 * 128 + k + kb + 1];
        b.x = W[(k + kb) * 128 + n0 + l16];
        b.y = W[(k + kb + 1) * 128 + n0 + l16];
        acc = __builtin_amdgcn_wmma_f32_16x16x4_f32(false, a, false, b, (short)0, acc, false, false);
    }
    #pragma unroll
    for (int v = 0; v < 8; ++v)
        h[(size_t)(r0 + v + half * 8) * 128 + n0 + l16] = acc[v];
}

// ---------- kernel 4: per-(node,head) attention coefficients ----------
__global__ __launch_bounds__(256) void k_attn_coef(
    const float* __restrict__ h, const float* __restrict__ a_src,
    const float* __restrict__ a_dst, float* __restrict__ as_, float* __restrict__ ad_)
{
    int idx = blockIdx.x * blockDim.x + threadIdx.x;
    if (idx >= N_NODES * H_NUM) return;
    int n = idx >> 3, hh = idx & 7;
    const float* hp = h + (size_t)n * 128 + hh * C_DIM;
    const float* s = a_src + hh * C_DIM;
    const float* d = a_dst + hh * C_DIM;
    float vs = 0.f, vd = 0.f;
    #pragma unroll
    for (int c = 0; c < C_DIM; ++c) { float v = hp[c]; vs += v * s[c]; vd += v * d[c]; }
    as_[idx] = vs; ad_[idx] = vd;
}

// ---------- kernel 5: edge logits + segment max ----------
__global__ __launch_bounds__(256) void k_edge_logit(
    const float* __restrict__ edge_attr, const int* __restrict__ src,
    const int* __restrict__ dst, const int* __restrict__ etype,
    const float* __restrict__ Wae, const float* __restrict__ type_ae,
    const float* __restrict__ as_, const float* __restrict__ ad_,
    float* __restrict__ logit, unsigned* __restrict__ mx)
{
    __shared__ float sW[ED_DIM * H_NUM];
    for (int i = threadIdx.x; i < ED_DIM * H_NUM; i += 256) sW[i] = Wae[i];
    __syncthreads();
    int e = blockIdx.x * blockDim.x + threadIdx.x;
    if (e >= N_EDGES) return;
    int s = src[e], d = dst[e], t = etype[e];
    float ea[ED_DIM];
    const float4* eap = (const float4*)(edge_attr + (size_t)e * ED_DIM);
    #pragma unroll
    for (int q = 0; q < 8; ++q) {
        float4 v = eap[q];
        ea[q * 4] = v.x; ea[q * 4 + 1] = v.y; ea[q * 4 + 2] = v.z; ea[q * 4 + 3] = v.w;
    }
    #pragma unroll
    for (int hh = 0; hh < H_NUM; ++hh) {
        float ae = type_ae[t * H_NUM + hh];
        #pragma unroll
        for (int k = 0; k < ED_DIM; ++k) ae += ea[k] * sW[k * H_NUM + hh];
        float lg = as_[s * H_NUM + hh] + ad_[d * H_NUM + hh] + ae;
        lg = (lg > 0.f) ? lg : 0.2f * lg;   // leaky_relu(0.2)
        logit[(size_t)e * H_NUM + hh] = lg;
        atomicMax(&mx[d * H_NUM + hh], f2ord(lg));
    }
}

// ---------- kernel 6: exp(logit - max) + segment sum of denom ----------
__global__ __launch_bounds__(256) void k_edge_softmax(
    float* __restrict__ logit, const int* __restrict__ dst,
    const unsigned* __restrict__ mx, float* __restrict__ denom)
{
    int idx = blockIdx.x * blockDim.x + threadIdx.x;
    if (idx >= N_EDGES * H_NUM) return;
    int e = idx >> 3, hh = idx & 7;
    int d = dst[e];
    float p = __expf(logit[idx] - ord2f(mx[d * H_NUM + hh]));
    logit[idx] = p;                       // reuse buffer as p
    atomicAdd(&denom[d * H_NUM + hh], p);
}

// ---------- kernel 7: GAT message scatter (4 features / thread) ----------
__global__ __launch_bounds__(256) void k_gat_scatter(
    const float* __restrict__ p, const float* __restrict__ denom,
    const int* __restrict__ src, const int* __restrict__ dst,
    const float* __restrict__ h, float* __restrict__ gat_acc)
{
    long long idx = (long long)blockIdx.x * blockDim.x + threadIdx.x; // over E*32
    if (idx >= (long long)N_EDGES * 32) return;
    int e = (int)(idx >> 5);
    int j0 = (int)(idx & 31) * 4;          // feature group of 4 (within one head)
    int hh = j0 >> 4;
    int s = src[e], d = dst[e];
    float alpha = p[(size_t)e * H_NUM + hh] / (denom[d * H_NUM + hh] + 1e-16f);
    float4 hv = *(const float4*)(h + (size_t)s * 128 + j0);
    float* dp = gat_acc + (size_t)d * 128 + j0;
    atomicAdd(dp + 0, alpha * hv.x);
    atomicAdd(dp + 1, alpha * hv.y);
    atomicAdd(dp + 2, alpha * hv.z);
    atomicAdd(dp + 3, alpha * hv.w);
}

// ---------- kernel 8: GINE edge GEMM (WMMA) + gather + relu + scatter ----------
__global__ __launch_bounds__(256) void k_gine_edge(
    const float* __restrict__ edge_attr, const int* __restrict__ src,
    const int* __restrict__ dst, const int* __restrict__ etype,
    const float* __restrict__ geW, const float* __restrict__ type_ep,
    const float* __restrict__ geb, const float* __restrict__ x,
    float* __restrict__ gine_agg)
{
    __shared__ float As[16 * ED_DIM];      // 2 KB, contiguous tile in global
    int e0 = blockIdx.x * 16;
    int tid = threadIdx.x;
    if (tid < 128) {
        unsigned lb = (unsigned)(uintptr_t)As;
        const char* gb = (const char*)(edge_attr + (size_t)e0 * ED_DIM);
        async_ld_b128(lb + tid * 16, gb + tid * 16);
    }
    async_wait0();
    __syncthreads();
    int wave = tid >> 5, lane = tid & 31;
    int half = lane >> 4, l16 = lane & 15, kb = half * 2;
    int n0 = wave * 16;
    v8f acc = {};
    #pragma unroll
    for (int k = 0; k < ED_DIM; k += 4) {
        v2f a, b;
        a.x = As[l16 * ED_DIM + k + kb];
        a.y = As[l16 * ED_DIM + k + kb + 1];
        b.x = geW[(k + kb) * IN_DIM + n0 + l16];
        b.y = geW[(k + kb + 1) * IN_DIM + n0 + l16];
        acc = __builtin_amdgcn_wmma_f32_16x16x4_f32(false, a, false, b, (short)0, acc, false, false);
    }
    int col = n0 + l16;
    float bn = geb[col];
    #pragma unroll
    for (int v = 0; v < 8; ++v) {
        int er = e0 + v + half * 8;
        int t = etype[er];
        float m = x[(size_t)src[er] * IN_DIM + col] + acc[v] + type_ep[t * IN_DIM + col] + bn;
        m = m > 0.f ? m : 0.f;
        atomicAdd(&gine_agg[(size_t)dst[er] * IN_DIM + col], m);
    }
}

// ---------- kernel 9: hid = relu(((1+eps)x + agg) @ W1 + b1) ----------
__global__ __launch_bounds__(256) void k_gine_mlp1(
    const float* __restrict__ x, const float* __restrict__ agg,
    const float* __restrict__ eps, const float* __restrict__ W1,
    const float* __restrict__ b1, float* __restrict__ hid)
{
    __shared__ float As[16 * 128];
    int r0 = blockIdx.x * 16;
    float ep1 = 1.0f + eps[0];
    int tid = threadIdx.x;
    for (int i = tid; i < 16 * 128; i += 256) {
        size_t gi = (size_t)r0 * 128 + i;
        As[i] = ep1 * x[gi] + agg[gi];
    }
    __syncthreads();
    int wave = tid >> 5, lane = tid & 31;
    int half = lane >> 4, l16 = lane & 15, kb = half * 2;
    int n0 = wave * 16;
    v8f acc = {};
    #pragma unroll 4
    for (int k = 0; k < 128; k += 4) {
        v2f a, b;
        a.x = As[l16 * 128 + k + kb];
        a.y = As[l16 * 128 + k + kb + 1];
        b.x = W1[(k + kb) * GINEC + n0 + l16];
        b.y = W1[(k + kb + 1) * GINEC + n0 + l16];
        acc = __builtin_amdgcn_wmma_f32_16x16x4_f32(false, a, false, b, (short)0, acc, false, false);
    }
    int col = n0 + l16;
    float bb = b1[col];
    #pragma unroll
    for (int v = 0; v < 8; ++v) {
        float z = acc[v] + bb;
        hid[(size_t)(r0 + v + half * 8) * GINEC + col] = z > 0.f ? z : 0.f;
    }
}

// ---------- kernel 10: gine_out = hid @ W2 + b2 ----------
__global__ __launch_bounds__(256) void k_gine_mlp2(
    const float* __restrict__ hid, const float* __restrict__ W2,
    const float* __restrict__ b2, float* __restrict__ gout)
{
    __shared__ float As[16 * 128];
    int r0 = blockIdx.x * 16;
    int tid = threadIdx.x;
    {
        unsigned lb = (unsigned)(uintptr_t)As;
        const char* gb = (const char*)(hid + (size_t)r0 * 128);
        async_ld_b128(lb + tid * 16, gb + tid * 16);
        async_ld_b128(lb + 4096 + tid * 16, gb + 4096 + tid * 16);
        async_wait0();
    }
    __syncthreads();
    int wave = tid >> 5, lane = tid & 31;
    int half = lane >> 4, l16 = lane & 15, kb = half * 2;
    int n0 = wave * 16;
    v8f acc = {};
    #pragma unroll 4
    for (int k = 0; k < 128; k += 4) {
        v2f a, b;
        a.x = As[l16 * 128 + k + kb];
        a.y = As[l16 * 128 + k + kb + 1];
        b.x = W2[(k + kb) * GINEC + n0 + l16];
        b.y = W2[(k + kb + 1) * GINEC + n0 + l16];
        acc = __builtin_amdgcn_wmma_f32_16x16x4_f32(false, a, false, b, (short)0, acc, false, false);
    }
    int col = n0 + l16;
    float bb = b2[col];
    #pragma unroll
    for (int v = 0; v < 8; ++v)
        gout[(size_t)(r0 + v + half * 8) * GINEC + col] = acc[v] + bb;
}

// ---------- kernel 11: z = concat @ comb_W + cb2 -> LayerNorm -> ReLU ----------
__global__ __launch_bounds__(256) void k_final(
    const float* __restrict__ gat_acc, const float* __restrict__ gout,
    const float* __restrict__ comb_W, const float* __restrict__ cb2,
    const float* __restrict__ ln_g, const float* __restrict__ ln_b,
    float* __restrict__ out)
{
    __shared__ float As[16 * 256];         // concat feature tile, 16 KB
    __shared__ float Zs[16 * 256];
    __shared__ float srow[32];             // mu, rsig per row
    int r0 = blockIdx.x * 16;
    int tid = threadIdx.x;
    {   // async-stage both halves raw (gat_b folded into cb2)
        unsigned lb = (unsigned)(uintptr_t)As;
        for (int i = tid; i < 512; i += 256) {       // 512 chunks of 16B per half
            int r = i >> 5, c = i & 31;              // 32 x 16B chunks per 128-f row
            async_ld_b128(lb + r * 1024 + c * 16,
                          gat_acc + (size_t)(r0 + r) * GATC + c * 4);
            async_ld_b128(lb + r * 1024 + 512 + c * 16,
                          gout + (size_t)(r0 + r) * GINEC + c * 4);
        }
        async_wait0();
    }
    __syncthreads();
    int wave = tid >> 5, lane = tid & 31;
    int half = lane >> 4, l16 = lane & 15, kb = half * 2;
    #pragma unroll
    for (int tcol = 0; tcol < 2; ++tcol) {
        int n0 = (wave * 2 + tcol) * 16;
        v8f acc = {};
        #pragma unroll 4
        for (int k = 0; k < 256; k += 4) {
            v2f a, b;
            a.x = As[l16 * 256 + k + kb];
            a.y = As[l16 * 256 + k + kb + 1];
            b.x = comb_W[(k + kb) * OUT_DIM + n0 + l16];
            b.y = comb_W[(k + kb + 1) * OUT_DIM + n0 + l16];
            acc = __builtin_amdgcn_wmma_f32_16x16x4_f32(false, a, false, b, (short)0, acc, false, false);
        }
        int col = n0 + l16;
        float bb = cb2[col];
        #pragma unroll
        for (int v = 0; v < 8; ++v)
            Zs[(v + half * 8) * 256 + col] = acc[v] + bb;
    }
    __syncthreads();
    if (tid < 16) {
        float s = 0.f, s2 = 0.f;
        const float* zr = &Zs[tid * 256];
        for (int j = 0; j < 256; ++j) { float v = zr[j]; s += v; s2 += v * v; }
        float mu = s * (1.f / 256.f);
        float var = s2 * (1.f / 256.f) - mu * mu;
        srow[tid * 2] = mu;
        srow[tid * 2 + 1] = rsqrtf(var + 1e-5f);
    }
    __syncthreads();
    for (int i = tid; i < 16 * 256; i += 256) {
        int r = i >> 8, j = i & 255;
        float v = (Zs[i] - srow[r * 2]) * srow[r * 2 + 1] * ln_g[j] + ln_b[j];
        out[(size_t)(r0 + r) * OUT_DIM + j] = v > 0.f ? v : 0.f;
    }
}

// ---------- host ----------
extern "C" void kernel_launch(void* const* d_in, const int* in_sizes, int n_in,
                              void* d_out, int out_size, void* d_ws, size_t ws_size,
                              hipStream_t stream) {
    const float* x            = (const float*)d_in[0];
    const int*   edge_index   = (const int*)d_in[1];
    const float* edge_attr    = (const float*)d_in[2];
    const int*   edge_types   = (const int*)d_in[3];
    const float* gat_W        = (const float*)d_in[4];
    const float* gat_b        = (const float*)d_in[5];
    const float* gat_edge_W   = (const float*)d_in[6];
    const float* gat_a_src    = (const float*)d_in[7];
    const float* gat_a_dst    = (const float*)d_in[8];
    const float* gat_a_edge   = (const float*)d_in[9];
    const float* gat_type_emb = (const float*)d_in[10];
    const float* gine_type_emb= (const float*)d_in[11];
    const float* gine_edge_W  = (const float*)d_in[12];
    const float* gine_edge_b  = (const float*)d_in[13];
    const float* gine_eps     = (const float*)d_in[14];
    const float* gine_W1      = (const float*)d_in[15];
    const float* gine_b1      = (const float*)d_in[16];
    const float* gine_W2      = (const float*)d_in[17];
    const float* gine_b2      = (const float*)d_in[18];
    const float* comb_W       = (const float*)d_in[19];
    const float* comb_b       = (const float*)d_in[20];
    const float* ln_g         = (const float*)d_in[21];
    const float* ln_b         = (const float*)d_in[22];

    const int* src = edge_index;
    const int* dst = edge_index + N_EDGES;

    float* ws = (float*)d_ws;
    size_t off = 0;
    float* h_buf    = ws + off; off += (size_t)N_NODES * 128;
    float* a_s      = ws + off; off += (size_t)N_NODES * H_NUM;
    float* a_d      = ws + off; off += (size_t)N_NODES * H_NUM;
    float* logit    = ws + off; off += (size_t)N_EDGES * H_NUM;
    unsigned* mx    = (unsigned*)(ws + off); off += (size_t)N_NODES * H_NUM;
    float* denom    = ws + off; off += (size_t)N_NODES * H_NUM;
    float* gat_acc  = ws + off; off += (size_t)N_NODES * 128;
    float* gine_agg = ws + off; off += (size_t)N_NODES * 128;
    float* hid      = ws + off; off += (size_t)N_NODES * 128;
    float* gout     = ws + off; off += (size_t)N_NODES * 128;
    float* Wae      = ws + off; off += 256;
    float* type_ae  = ws + off; off += 128;
    float* type_ep  = ws + off; off += 2048;
    float* cb2      = ws + off; off += 256;

    k_precompute<<<1, 256, 0, stream>>>(gat_edge_W, gat_a_edge, gat_type_emb,
                                        gine_type_emb, gine_edge_W,
                                        gat_b, comb_W, comb_b,
                                        Wae, type_ae, type_ep, cb2);
    k_init<<<4096, 256, 0, stream>>>(mx, denom, gat_acc, gine_agg,
                                     N_NODES * H_NUM, N_NODES * 128);
    k_gemm_h<<<N_NODES / 16, 256, 0, stream>>>(x, gat_W, h_buf);
    k_attn_coef<<<(N_NODES * H_NUM + 255) / 256, 256, 0, stream>>>(h_buf, gat_a_src, gat_a_dst, a_s, a_d);
    k_edge_logit<<<(N_EDGES + 255) / 256, 256, 0, stream>>>(edge_attr, src, dst, edge_types,
                                                            Wae, type_ae, a_s, a_d, logit, mx);
    k_edge_softmax<<<(N_EDGES * H_NUM + 255) / 256, 256, 0, stream>>>(logit, dst, mx, denom);
    k_gat_scatter<<<(int)(((long long)N_EDGES * 32 + 255) / 256), 256, 0, stream>>>(
        logit, denom, src, dst, h_buf, gat_acc);
    k_gine_edge<<<N_EDGES / 16, 256, 0, stream>>>(edge_attr, src, dst, edge_types,
                                                  gine_edge_W, type_ep, gine_edge_b, x, gine_agg);
    k_gine_mlp1<<<N_NODES / 16, 256, 0, stream>>>(x, gine_agg, gine_eps, gine_W1, gine_b1, hid);
    k_gine_mlp2<<<N_NODES / 16, 256, 0, stream>>>(hid, gine_W2, gine_b2, gout);
    k_final<<<N_NODES / 16, 256, 0, stream>>>(gat_acc, gout, comb_W, cb2,
                                              ln_g, ln_b, (float*)d_out);
}